// WayfinderAttentionMLX_66992899883625
// MI455X (gfx1250) — compile-verified
//
#include <hip/hip_runtime.h>

// Problem constants (match reference)
#define B_    2
#define H_    8
#define T_    2048
#define DH_   64
#define WIN_  32
#define WTOT  80            // gathered window rows per 16-query tile: 16 + 2*32
#define KSTR  68            // LDS row stride (floats): 272B rows -> 16B aligned, banks 4*row mod 64
#define WSTR  84            // LDS row stride for weight tile: banks 20*m mod 64, conflict-free

typedef float v2f __attribute__((ext_vector_type(2)));
typedef float v8f __attribute__((ext_vector_type(8)));

// Async copy: 16 bytes per lane, global -> LDS, tracked by ASYNCcnt (CDNA5).
__device__ __forceinline__ void async_b128(unsigned lds_off, const float* g) {
    asm volatile("global_load_async_to_lds_b128 %0, %1, off"
                 :: "v"(lds_off), "v"(g) : "memory");
}

__global__ __launch_bounds__(32)
void wayfinder_attn_wmma(const float* __restrict__ q,
                         const float* __restrict__ k,
                         const float* __restrict__ v,
                         const int*   __restrict__ perms,
                         float*       __restrict__ out)
{
    __shared__ __attribute__((aligned(16))) float Ksh[WTOT][KSTR];
    __shared__ __attribute__((aligned(16))) float Vsh[WTOT][KSTR];
    __shared__ __attribute__((aligned(16))) float Qsh[16][KSTR];
    __shared__ __attribute__((aligned(16))) float Wsh[16][WSTR];
    __shared__ int   permw[WTOT];

    const int lane  = threadIdx.x;     // 0..31 (wave32)
    const int half  = lane >> 4;       // 0 / 1
    const int ln    = lane & 15;
    const int tiles = T_ / 16;
    const int bid   = blockIdx.x;
    const int bh    = bid / tiles;     // fused (b,h)
    const int qt    = bid % tiles;
    const int h     = bh % H_;
    const int i0    = qt * 16;         // first permuted query index of this tile

    const int* ph = perms + h * T_;

    // ---- stage window permutation values (clamped positions) ----
    for (int j = lane; j < WTOT; j += 32) {
        int p = i0 - WIN_ + j;
        p = p < 0 ? 0 : (p >= T_ ? T_ - 1 : p);
        permw[j] = ph[p];
    }
    __syncthreads();

    // ---- async gather of K/V window rows + Q tile rows straight into LDS ----
    // 2 rows per instruction: lanes 0-15 cover row rr, lanes 16-31 row rr+1,
    // 16 bytes per lane. Addresses are per-lane 64-bit (GV mode).
    const float* kb = k + (size_t)bh * T_ * DH_;
    const float* vb = v + (size_t)bh * T_ * DH_;
    const float* qb = q + (size_t)bh * T_ * DH_;
    for (int rr = 0; rr < WTOT; rr += 2) {
        int row  = rr + half;
        int orig = permw[row];
        async_b128((unsigned)(size_t)&Ksh[row][ln * 4], kb + (size_t)orig * DH_ + ln * 4);
        async_b128((unsigned)(size_t)&Vsh[row][ln * 4], vb + (size_t)orig * DH_ + ln * 4);
    }
    for (int rr = 0; rr < 16; rr += 2) {
        int row  = rr + half;
        int orig = permw[WIN_ + row];
        async_b128((unsigned)(size_t)&Qsh[row][ln * 4], qb + (size_t)orig * DH_ + ln * 4);
    }
    asm volatile("s_wait_asynccnt 0x0" ::: "memory");
    __syncthreads();

    // ---- preload Q A-operands (A layout: lane ln = row, K-pair 2*half) ----
    v2f qa[16];
#pragma unroll
    for (int c = 0; c < 16; ++c) {
        qa[c].x = Qsh[ln][4 * c + 2 * half];
        qa[c].y = Qsh[ln][4 * c + 2 * half + 1];
    }

    // ---- S = Q * K^T : 5 key sub-tiles, 16 WMMA f32 16x16x4 each ----
    v8f acc[5];
#pragma unroll
    for (int t = 0; t < 5; ++t) {
        v8f c8 = {0.f, 0.f, 0.f, 0.f, 0.f, 0.f, 0.f, 0.f};
#pragma unroll
        for (int c = 0; c < 16; ++c) {
            v2f bop;   // B[k][n] = K[16t+n][4c+k]; lane holds col n=ln, K-pair 2*half
            bop.x = Ksh[16 * t + ln][4 * c + 2 * half];
            bop.y = Ksh[16 * t + ln][4 * c + 2 * half + 1];
            c8 = __builtin_amdgcn_wmma_f32_16x16x4_f32(
                     false, qa[c], false, bop, (short)0, c8, false, false);
        }
        acc[t] = c8;
    }

    // ---- masked softmax over the 80-wide window (65 valid per row) ----
    const float scale = 0.125f;   // 1/sqrt(64)
#pragma unroll
    for (int r = 0; r < 8; ++r) {
        const int mrow = r + 8 * half;          // C layout: VGPR r, half selects M
        const int oq   = permw[WIN_ + mrow];    // orig index of this query row
        float sv[5];
        int   vmask = 0;
        float mx = -1e30f;
#pragma unroll
        for (int t = 0; t < 5; ++t) {
            int j = 16 * t + ln;                // window column
            int p = i0 - WIN_ + j;              // permuted key position
            int w = j - WIN_ - mrow;            // offset
            bool valid = (p >= 0) && (p < T_) && (w >= -WIN_) && (w <= WIN_)
                         && (permw[j] <= oq);
            float s = acc[t][r] * scale + ((w == 1 || w == -1) ? 0.5f : 0.0f);
            s = valid ? s : -1e30f;
            if (valid) vmask |= (1 << t);
            sv[t] = s;
            mx = fmaxf(mx, s);
        }
#pragma unroll
        for (int off = 1; off < 16; off <<= 1)
            mx = fmaxf(mx, __shfl_xor(mx, off, 16));
        float sum = 0.f;
        float ev[5];
#pragma unroll
        for (int t = 0; t < 5; ++t) {
            float e = (vmask >> t) & 1 ? __expf(sv[t] - mx) : 0.0f;
            ev[t] = e;
            sum += e;
        }
#pragma unroll
        for (int off = 1; off < 16; off <<= 1)
            sum += __shfl_xor(sum, off, 16);
        float inv = 1.0f / fmaxf(sum, 1e-9f);
#pragma unroll
        for (int t = 0; t < 5; ++t)
            Wsh[mrow][16 * t + ln] = ev[t] * inv;
    }
    __syncthreads();

    // ---- preload weight A-operands (20 chunks of K=4 over 80 keys) ----
    v2f wa[20];
#pragma unroll
    for (int c = 0; c < 20; ++c) {
        wa[c].x = Wsh[ln][4 * c + 2 * half];
        wa[c].y = Wsh[ln][4 * c + 2 * half + 1];
    }

    // ---- Y = W * V : 4 output d-tiles, 20 WMMA each; scatter to y[perm[i]] ----
    float* ob = out + (size_t)bh * T_ * DH_;
#pragma unroll
    for (int nt = 0; nt < 4; ++nt) {
        v8f o8 = {0.f, 0.f, 0.f, 0.f, 0.f, 0.f, 0.f, 0.f};
#pragma unroll
        for (int c = 0; c < 20; ++c) {
            v2f bop;   // B[k][n] = V[4c+k][16nt+n]
            bop.x = Vsh[4 * c + 2 * half][16 * nt + ln];
            bop.y = Vsh[4 * c + 2 * half + 1][16 * nt + ln];
            o8 = __builtin_amdgcn_wmma_f32_16x16x4_f32(
                     false, wa[c], false, bop, (short)0, o8, false, false);
        }
#pragma unroll
        for (int r = 0; r < 8; ++r) {
            int mrow = r + 8 * half;
            int orig = permw[WIN_ + mrow];      // y[perm[i]] = y_p[i]
            ob[(size_t)orig * DH_ + 16 * nt + ln] = o8[r];
        }
    }
}

extern "C" void kernel_launch(void* const* d_in, const int* in_sizes, int n_in,
                              void* d_out, int out_size, void* d_ws, size_t ws_size,
                              hipStream_t stream) {
    (void)in_sizes; (void)n_in; (void)out_size; (void)d_ws; (void)ws_size;
    const float* q     = (const float*)d_in[0];
    const float* k     = (const float*)d_in[1];
    const float* v     = (const float*)d_in[2];
    const int*   perms = (const int*)d_in[3];
    // d_in[4] (inverse perms) not needed: we scatter rows via perms directly.
    dim3 grid(B_ * H_ * (T_ / 16));
    wayfinder_attn_wmma<<<grid, 32, 0, stream>>>(q, k, v, perms, (float*)d_out);
}